// DiGCN_link_prediction_50491635532107
// MI455X (gfx1250) — compile-verified
//
#include <hip/hip_runtime.h>
#include <hip/hip_bf16.h>

// ---------------------------------------------------------------------------
// DiGCN link prediction for MI455X (gfx1250, wave32).
//   T  = X @ W      WMMA f32 16x16x4, exact fp32.
//                   Register-blocked: one wave32 computes a 16x64 strip
//                   (4 accumulators) so each A fragment feeds 4 WMMAs.
//   H  = b (broadcast); scatter-add over edges via global_atomic_add_f32.
//                   Node features (50000x256 f32 = 51.2MB) fit in the 192MB
//                   L2, so the random edge gather/scatter stays on-chip.
//   relu fused into the 2nd GEMM's A-read (saves a 51MB pass).
//   head: one wave32 per query, shuffle reduction, log_softmax over 2 logits.
// ---------------------------------------------------------------------------

#define N_NODES 50000
#define N_EDGES 1000000
#define N_QUERY 100000
#define FDIM    256

typedef __attribute__((ext_vector_type(2))) float v2f;
typedef __attribute__((ext_vector_type(8))) float v8f;

__device__ __forceinline__ v8f wmma_f32(v2f a, v2f b, v8f c) {
    // (neg_a, A, neg_b, B, c_mod, C, reuse_a, reuse_b)
    return __builtin_amdgcn_wmma_f32_16x16x4_f32(
        false, a, false, b, (short)0, c, false, false);
}

// ---------------------------------------------------------------------------
// C[M,256] = op(A[M,256]) @ W[256,256], op = identity or relu (fused).
// Fragment layouts per CDNA5 ISA 7.12.2:
//   A 16x4:  lanes 0-15 hold (K,K+1) of row M=lane; lanes 16-31 hold (K+2,K+3)
//   B 4x16:  lanes 0-15 hold rows K,K+1 at col N=lane; lanes 16-31 rows K+2,K+3
//   D 16x16: VGPR r = row r (lanes 0-15) / row r+8 (lanes 16-31), col = lane%16
// Block = 256 threads = 8 waves: 2 M-tiles x 4 N-strips of 64 columns.
// ---------------------------------------------------------------------------
template <bool RELU>
__global__ __launch_bounds__(256) void gemm_wmma_f32(
    const float* __restrict__ A,   // [M, 256] row-major
    const float* __restrict__ W,   // [256, 256] row-major
    float* __restrict__ C,         // [M, 256] row-major
    int M)
{
    const int lane  = threadIdx.x & 31;
    const int waveI = threadIdx.x >> 5;                 // 0..7
    const int tm    = (blockIdx.x * 2 + (waveI >> 2)) * 16;
    const int tn    = (waveI & 3) * 64;                 // 64-wide N strip
    if (tm >= M) return;                                // wave-uniform

    const int half = lane >> 4;                         // 0: lanes 0-15, 1: 16-31
    const int l16  = lane & 15;

    const float* __restrict__ arow = A + (size_t)(tm + l16) * FDIM;
    const float* __restrict__ bcol = W + tn + l16;

    v8f acc0 = {}, acc1 = {}, acc2 = {}, acc3 = {};
#pragma unroll 4
    for (int k0 = 0; k0 < FDIM; k0 += 4) {
        const int k = k0 + half * 2;
        float a0 = arow[k];
        float a1 = arow[k + 1];
        if (RELU) { a0 = fmaxf(a0, 0.0f); a1 = fmaxf(a1, 0.0f); }
        const v2f a = { a0, a1 };

        const float* __restrict__ br0 = bcol + (size_t)k * FDIM;
        const float* __restrict__ br1 = br0 + FDIM;
        const v2f b0 = { br0[ 0], br1[ 0] };
        const v2f b1 = { br0[16], br1[16] };
        const v2f b2 = { br0[32], br1[32] };
        const v2f b3 = { br0[48], br1[48] };

        acc0 = wmma_f32(a, b0, acc0);      // A fragment reused 4x
        acc1 = wmma_f32(a, b1, acc1);
        acc2 = wmma_f32(a, b2, acc2);
        acc3 = wmma_f32(a, b3, acc3);
    }

    float* __restrict__ crow = C + (size_t)(tm + half * 8) * FDIM + tn + l16;
#pragma unroll
    for (int r = 0; r < 8; ++r) {
        crow[(size_t)r * FDIM +  0] = acc0[r];
        crow[(size_t)r * FDIM + 16] = acc1[r];
        crow[(size_t)r * FDIM + 32] = acc2[r];
        crow[(size_t)r * FDIM + 48] = acc3[r];
    }
}

// H[n][f] = bias[f]  (scatter-add accumulates on top of the bias)
__global__ __launch_bounds__(256) void init_bias(
    float* __restrict__ H, const float* __restrict__ bias, int M)
{
    const int idx = blockIdx.x * 256 + threadIdx.x;
    if (idx < M * FDIM) H[idx] = bias[idx & (FDIM - 1)];
}

// One block per edge; thread f handles feature f.
// Gather T[src] (coalesced, L2-resident), scale by w_e, atomic-add into H[dst].
__global__ __launch_bounds__(256) void scatter_edges(
    const float* __restrict__ T,
    const int*   __restrict__ src,
    const int*   __restrict__ dst,
    const float* __restrict__ ew,
    float* __restrict__ H)
{
    const int e = blockIdx.x;
    const int f = threadIdx.x;
    const int   s = src[e];          // block-uniform -> scalar loads
    const int   d = dst[e];
    const float w = ew[e];
    const float v = T[(size_t)s * FDIM + f] * w;
    atomicAdd(&H[(size_t)d * FDIM + f], v);   // global_atomic_add_f32 at L2
}

// One wave32 per query: logits = concat(H2[a], H2[b]) @ Wl + bl; log_softmax.
__global__ __launch_bounds__(256) void link_head(
    const float* __restrict__ H2,
    const int*   __restrict__ q,      // [NQ, 2]
    const float* __restrict__ Wl,     // [512, 2]
    const float* __restrict__ bl,     // [2]
    float* __restrict__ out,          // [NQ, 2]
    int NQ)
{
    const int gtid = blockIdx.x * blockDim.x + threadIdx.x;
    const int wave = gtid >> 5;
    const int lane = gtid & 31;
    if (wave >= NQ) return;

    const int na = q[wave * 2 + 0];
    const int nb = q[wave * 2 + 1];
    const float* __restrict__ ha = H2 + (size_t)na * FDIM;
    const float* __restrict__ hb = H2 + (size_t)nb * FDIM;

    float s0 = 0.0f, s1 = 0.0f;
#pragma unroll
    for (int j = lane; j < FDIM; j += 32) {
        const float va = ha[j];
        const float vb = hb[j];
        s0 += va * Wl[j * 2 + 0] + vb * Wl[(j + FDIM) * 2 + 0];
        s1 += va * Wl[j * 2 + 1] + vb * Wl[(j + FDIM) * 2 + 1];
    }
#pragma unroll
    for (int off = 16; off > 0; off >>= 1) {
        s0 += __shfl_down(s0, off, 32);
        s1 += __shfl_down(s1, off, 32);
    }
    if (lane == 0) {
        const float l0 = s0 + bl[0];
        const float l1 = s1 + bl[1];
        const float m  = fmaxf(l0, l1);
        const float lse = m + __logf(__expf(l0 - m) + __expf(l1 - m));
        out[wave * 2 + 0] = l0 - lse;
        out[wave * 2 + 1] = l1 - lse;
    }
}

extern "C" void kernel_launch(void* const* d_in, const int* in_sizes, int n_in,
                              void* d_out, int out_size, void* d_ws, size_t ws_size,
                              hipStream_t stream) {
    const float* x   = (const float*)d_in[0];   // [50000, 256]
    const int*   ei  = (const int*)  d_in[1];   // [2, 1M] row-major
    const int*   qe  = (const int*)  d_in[2];   // [100000, 2]
    const float* ew  = (const float*)d_in[3];   // [1M]
    const float* W1  = (const float*)d_in[4];
    const float* b1  = (const float*)d_in[5];
    const float* W2  = (const float*)d_in[6];
    const float* b2  = (const float*)d_in[7];
    const float* Wl  = (const float*)d_in[8];   // [512, 2]
    const float* bl  = (const float*)d_in[9];   // [2]

    const int* src = ei;
    const int* dst = ei + N_EDGES;

    float* T  = (float*)d_ws;                      // 50000*256 f32 (51.2 MB)
    float* H1 = T  + (size_t)N_NODES * FDIM;
    float* H2 = H1 + (size_t)N_NODES * FDIM;       // total 153.6 MB workspace

    const int mTiles     = N_NODES / 16;                   // 3125
    const int gemmBlocks = (mTiles + 1) / 2;               // 2 M-tiles / block
    const int initBlocks = (N_NODES * FDIM + 255) / 256;
    const int headBlocks = (N_QUERY * 32 + 255) / 256;

    // Layer 1: T = X@W1 ; H1 = b1 + scatter(T)
    gemm_wmma_f32<false><<<gemmBlocks, 256, 0, stream>>>(x, W1, T, N_NODES);
    init_bias<<<initBlocks, 256, 0, stream>>>(H1, b1, N_NODES);
    scatter_edges<<<N_EDGES, 256, 0, stream>>>(T, src, dst, ew, H1);

    // Layer 2: T = relu(H1)@W2 (relu fused) ; H2 = b2 + scatter(T)
    gemm_wmma_f32<true><<<gemmBlocks, 256, 0, stream>>>(H1, W2, T, N_NODES);
    init_bias<<<initBlocks, 256, 0, stream>>>(H2, b2, N_NODES);
    scatter_edges<<<N_EDGES, 256, 0, stream>>>(T, src, dst, ew, H2);

    // Head
    link_head<<<headBlocks, 256, 0, stream>>>(H2, qe, Wl, bl, (float*)d_out, N_QUERY);
}